// MORAL_41308995452996
// MI455X (gfx1250) — compile-verified
//
#include <hip/hip_runtime.h>

#define CH 128
#define TILE_M 16
#define RB 200   // row-blocks for min/max reduction

typedef __attribute__((ext_vector_type(2))) float v2f;
typedef __attribute__((ext_vector_type(8))) float v8f;

// ---------------- column min/max (two-pass, deterministic) ----------------
__global__ __launch_bounds__(CH)
void colminmax_partial(const float* __restrict__ x, int nrows, int rows_per_blk,
                       float* __restrict__ pmin, float* __restrict__ pmax) {
  int c = threadIdx.x;                       // one thread per column
  int r0 = blockIdx.x * rows_per_blk;
  int r1 = r0 + rows_per_blk; if (r1 > nrows) r1 = nrows;
  float mn = 3.4e38f, mx = -3.4e38f;
  for (int r = r0; r < r1; ++r) {
    float v = x[(size_t)r * CH + c];
    mn = fminf(mn, v); mx = fmaxf(mx, v);
  }
  pmin[blockIdx.x * CH + c] = mn;
  pmax[blockIdx.x * CH + c] = mx;
}

__global__ __launch_bounds__(CH)
void colminmax_final(const float* __restrict__ pmin, const float* __restrict__ pmax,
                     int nblk, float* __restrict__ mn_out, float* __restrict__ inv_out) {
  int c = threadIdx.x;
  float mn = 3.4e38f, mx = -3.4e38f;
  for (int b = 0; b < nblk; ++b) {
    mn = fminf(mn, pmin[b * CH + c]);
    mx = fmaxf(mx, pmax[b * CH + c]);
  }
  float d = (mx > mn) ? (mx - mn) : 1.0f;
  mn_out[c]  = mn;
  inv_out[c] = 1.0f / d;
}

// ---------------- degree / dinv ----------------
__global__ void deg_init(float* __restrict__ deg, int n) {
  int i = blockIdx.x * 256 + threadIdx.x;
  if (i < n) deg[i] = 1.0f;                  // self-loop contributes 1
}
__global__ void deg_count(const int* __restrict__ dst, int E, float* __restrict__ deg) {
  int e = blockIdx.x * 256 + threadIdx.x;
  if (e < E) atomicAdd(&deg[dst[e]], 1.0f);
}
__global__ void deg_to_dinv(float* __restrict__ deg, int n) {
  int i = blockIdx.x * 256 + threadIdx.x;
  if (i < n) deg[i] = rsqrtf(deg[i]);        // deg >= 1 always
}

// ---------------- GEMM: H = A @ W with f32 WMMA (optional fused min-max norm) ----------------
// Block = 256 threads = 8 waves; block computes 16 rows x 128 cols of H.
// Wave w owns output columns [16w, 16w+16). K=128 -> 32 chained V_WMMA_F32_16X16X4_F32.
__global__ __launch_bounds__(256)
void gcn_gemm(const float* __restrict__ A, const float* __restrict__ W,
              float* __restrict__ H, int nrows,
              const float* __restrict__ mn, const float* __restrict__ inv, int do_norm) {
  __shared__ float As[TILE_M][CH + 4];       // pad: stride 132 % 64 banks = 4 -> conflict-free column reads
  int tid  = threadIdx.x;
  int row0 = blockIdx.x * TILE_M;

  // cooperative load (+ normalize) of the 16x128 A tile: 2048 elems, 8 per thread
  for (int i = 0; i < 8; ++i) {
    int idx = tid + 256 * i;
    int r = idx >> 7, c = idx & 127;
    float v = 0.0f;
    if (row0 + r < nrows) v = A[(size_t)(row0 + r) * CH + c];
    if (do_norm) v = (v - mn[c]) * inv[c];
    As[r][c] = v;
  }
  __syncthreads();

  int wave = tid >> 5, lane = tid & 31;
  int col0 = wave * 16;
  int half = lane >> 4;                      // lanes 0-15 carry K pair {0,1}; 16-31 carry {2,3}
  int l16  = lane & 15;

  v8f acc = {0.f, 0.f, 0.f, 0.f, 0.f, 0.f, 0.f, 0.f};
  for (int k0 = 0; k0 < CH; k0 += 4) {
    int kk = k0 + half * 2;
    v2f a, b;
    // A fragment (16x4, 32-bit): lane = M (0-15), VGPR0/1 = consecutive K, halves split K
    a.x = As[l16][kk];
    a.y = As[l16][kk + 1];
    // B fragment (4x16, 32-bit): lane = N within half, VGPR0/1 = consecutive K rows
    b.x = W[(size_t)kk * CH + col0 + l16];
    b.y = W[(size_t)(kk + 1) * CH + col0 + l16];
    acc = __builtin_amdgcn_wmma_f32_16x16x4_f32(false, a, false, b, (short)0, acc, false, false);
  }

  // D layout: VGPR r -> row r (lanes 0-15) / row 8+r (lanes 16-31), col = lane%16
  int rbase = half * 8;
  for (int r = 0; r < 8; ++r) {
    int orow = row0 + rbase + r;
    if (orow < nrows) H[(size_t)orow * CH + col0 + l16] = acc[r];
  }
}

// ---------------- aggregation ----------------
// out[n][c] = bias[c] + h[n][c] * dinv[n]^2   (self-loop term + bias, also zero-inits accumulator)
__global__ __launch_bounds__(256)
void agg_init(const float* __restrict__ h, const float* __restrict__ dinv,
              const float* __restrict__ bias, float* __restrict__ out, int total) {
  int idx = blockIdx.x * 256 + threadIdx.x;
  if (idx >= total) return;
  int node = idx >> 7, c = idx & 127;
  float di = dinv[node];
  out[idx] = bias[c] + h[idx] * di * di;
}

// one wave32 per edge: gather 128 floats (float4/lane), scale, 4 atomic f32 adds per lane
__global__ __launch_bounds__(256)
void agg_edges(const float* __restrict__ h, const float* __restrict__ dinv,
               const int* __restrict__ src, const int* __restrict__ dst,
               int E, float* __restrict__ out) {
  int e = blockIdx.x * 8 + (threadIdx.x >> 5);
  if (e >= E) return;
  int lane = threadIdx.x & 31;
  int s = src[e], d = dst[e];
  float norm = dinv[s] * dinv[d];
  const float4* hp = (const float4*)(h + (size_t)s * CH);
  float4 v = hp[lane];
  float* op = out + (size_t)d * CH + lane * 4;
  atomicAdd(op + 0, v.x * norm);
  atomicAdd(op + 1, v.y * norm);
  atomicAdd(op + 2, v.z * norm);
  atomicAdd(op + 3, v.w * norm);
}

__global__ void relu_ip(float* __restrict__ x, int n) {
  int i = blockIdx.x * 256 + threadIdx.x;
  if (i < n) x[i] = fmaxf(x[i], 0.0f);
}

// ---------------- driver ----------------
extern "C" void kernel_launch(void* const* d_in, const int* in_sizes, int n_in,
                              void* d_out, int out_size, void* d_ws, size_t ws_size,
                              hipStream_t stream) {
  const float* features = (const float*)d_in[0];   // [N,128]
  const float* W1       = (const float*)d_in[1];   // [128,128]
  const float* b1       = (const float*)d_in[2];   // [128]
  const float* W2       = (const float*)d_in[3];   // [128,128]
  const float* b2       = (const float*)d_in[4];   // [128]
  const int*   ei       = (const int*)d_in[5];     // [2,E] (template: integer -> const int*)
  // d_in[6] = group (scalar) selects identically-shaped params; unused here.

  const int N = in_sizes[0] / CH;
  const int E = in_sizes[5] / 2;
  const int* src = ei;
  const int* dst = ei + E;

  float* out = (float*)d_out;                      // [N,128] f32
  const int total = N * CH;

  // workspace layout (all f32)
  float* deg  = (float*)d_ws;                      // N       (becomes dinv in place)
  float* mn   = deg + N;                           // 128
  float* inv  = mn + CH;                           // 128
  float* pmin = inv + CH;                          // RB*128
  float* pmax = pmin + RB * CH;                    // RB*128
  float* h    = pmax + RB * CH;                    // N*128   (~51.2 MB)

  const int rows_per_blk = (N + RB - 1) / RB;
  const int mtiles   = (N + TILE_M - 1) / TILE_M;  // 6250
  const int eblk256  = (E + 255) / 256;
  const int nblk256  = (N + 255) / 256;
  const int tblk256  = (total + 255) / 256;
  const int edgeblk  = (E + 7) / 8;                // 8 edges (waves) per block

  // 1) column min/max + inverse range
  colminmax_partial<<<RB, CH, 0, stream>>>(features, N, rows_per_blk, pmin, pmax);
  colminmax_final<<<1, CH, 0, stream>>>(pmin, pmax, RB, mn, inv);

  // 2) degrees (with self-loops) -> dinv
  deg_init<<<nblk256, 256, 0, stream>>>(deg, N);
  deg_count<<<eblk256, 256, 0, stream>>>(dst, E, deg);
  deg_to_dinv<<<nblk256, 256, 0, stream>>>(deg, N);

  // 3) layer 1: h = normalize(x) @ W1  (WMMA f32, norm fused into tile load)
  gcn_gemm<<<mtiles, 256, 0, stream>>>(features, W1, h, N, mn, inv, 1);

  // 4) aggregate: out = b1 + selfloop + scatter(edges)
  agg_init<<<tblk256, 256, 0, stream>>>(h, deg, b1, out, total);
  agg_edges<<<edgeblk, 256, 0, stream>>>(h, deg, src, dst, E, out);

  // 5) ReLU in place
  relu_ip<<<tblk256, 256, 0, stream>>>(out, total);

  // 6) layer 2: h = out @ W2 (GEMM reads out before it is overwritten; stream-ordered)
  gcn_gemm<<<mtiles, 256, 0, stream>>>(out, W2, h, N, mn, inv, 0);

  // 7) aggregate into d_out
  agg_init<<<tblk256, 256, 0, stream>>>(h, deg, b2, out, total);
  agg_edges<<<edgeblk, 256, 0, stream>>>(h, deg, src, dst, E, out);
}